// QMLP_38663295598914
// MI455X (gfx1250) — compile-verified
//
#include <hip/hip_runtime.h>
#include <cstdint>

// Quaternion MLP: two block-structured fp32 GEMMs on the WMMA f32 16x16x4 path.
//   h   = relu(Xcat @ W1_block^T + b1cat)      (stored in d_ws, [N, 2048])
//   out = h @ W2_block^T + b2cat               (component-major in d_out)
// 4x4 sign-structured block weights applied on the fly: (ic,oc) selects the
// raw 512x512 weight array; the +-1 sign is folded into the A fragments at
// compute time so the LDS fill can be a pure async copy.

#define DCOMP 512
#define NROWS 32768
#define KTOT  2048
#define M_BLK 128
#define N_BLK 128
#define KT    16
#define LDA   20        // padded LDS row stride (floats); keeps 16B alignment
#define NKT   (KTOT / KT)

#define AS1 __attribute__((address_space(1)))
#define AS3 __attribute__((address_space(3)))

#if __has_builtin(__builtin_amdgcn_global_load_async_to_lds_b128) && \
    __has_builtin(__builtin_amdgcn_s_wait_asynccnt)
#define HAVE_ASYNC 1
#else
#define HAVE_ASYNC 0
#endif

typedef __attribute__((ext_vector_type(2))) float v2f;
typedef __attribute__((ext_vector_type(4))) float v4f;
typedef __attribute__((ext_vector_type(8))) float v8f;
typedef __attribute__((ext_vector_type(4))) int   v4i;

// W_block[oc*512+oo, ic*512+ii] = c_sgn[ic][oc] * W_{c_midx[ic][oc]}[oo, ii]
__device__ __constant__ int   c_midx[4][4] = {
    {0, 1, 2, 3},
    {1, 0, 3, 2},
    {2, 3, 0, 1},
    {3, 2, 1, 0}};
__device__ __constant__ float c_sgn[4][4] = {
    { 1.f,  1.f,  1.f,  1.f},
    {-1.f,  1.f, -1.f,  1.f},
    {-1.f,  1.f,  1.f, -1.f},
    {-1.f, -1.f,  1.f,  1.f}};

// ---- address-space helpers (integer round-trip: always-legal casts; the
// generic LDS pointer's low 32 bits are the AS3 offset per the aperture) ----
__device__ __forceinline__ v4f gload4(const float* p) {
    return *(const AS1 v4f*)(uintptr_t)p;
}
__device__ __forceinline__ float gloadf(const float* p) {
    return *(const AS1 float*)(uintptr_t)p;
}
__device__ __forceinline__ void gstoref(float* p, float v) {
    *(AS1 float*)(uintptr_t)p = v;
}
#if HAVE_ASYNC
__device__ __forceinline__ void async_g2l_16B(const float* g, float* l) {
    __builtin_amdgcn_global_load_async_to_lds_b128(
        (AS1 v4i*)(uintptr_t)g, (AS3 v4i*)(uint32_t)(uintptr_t)l,
        /*imm offset*/ 0, /*cpol*/ 0);
}
#endif

__device__ __forceinline__ v8f wmma4(v2f a, v2f b, v8f c) {
    return __builtin_amdgcn_wmma_f32_16x16x4_f32(
        false, a, false, b, (short)0, c, false, false);
}

// LAYER==1: A = 4 q arrays (stride 512, component-selected), ReLU, Out = h.
// LAYER==2: A = h (stride 2048), no ReLU, Out component-major out[oc][n][512].
template <int LAYER>
__global__ __launch_bounds__(256) void qmlp_gemm(
    const float* __restrict__ A0, const float* __restrict__ A1,
    const float* __restrict__ A2, const float* __restrict__ A3,
    const float* __restrict__ W0, const float* __restrict__ W1,
    const float* __restrict__ W2, const float* __restrict__ W3,
    const float* __restrict__ B0, const float* __restrict__ B1,
    const float* __restrict__ B2, const float* __restrict__ B3,
    float* __restrict__ Out) {
    const int tid  = threadIdx.x;
    const int lane = tid & 31;          // wave32
    const int wid  = tid >> 5;          // 8 waves per block
    const int wm   = wid & 3;           // 4 waves along M (32 rows each)
    const int wn   = wid >> 2;          // 2 waves along N (64 cols each)
    const int m0   = blockIdx.y * M_BLK;
    const int nb   = blockIdx.x;        // 0..15 over 2048 output features
    const int oc   = nb >> 2;           // output component
    const int n0r  = (nb & 3) * N_BLK;  // column offset within component

    __shared__ float As[2][M_BLK * LDA];
    __shared__ float Bs[2][N_BLK * LDA];

    const float* Aq[4] = {A0, A1, A2, A3};
    const float* Wv[4] = {W0, W1, W2, W3};
    const float* Bv[4] = {B0, B1, B2, B3};

    // loader mapping: 4 threads x 16B cover one 16-float row; rows lr, lr+64.
    const int lr = tid >> 2;
    const int lc = (tid & 3) << 2;

    v8f acc[2][4];
#pragma unroll
    for (int i = 0; i < 2; ++i)
#pragma unroll
        for (int j = 0; j < 4; ++j)
            acc[i][j] = (v8f){0.f, 0.f, 0.f, 0.f, 0.f, 0.f, 0.f, 0.f};

    // source addresses for K-chunk kt
    auto srcA = [&](int kt, int rowoff) -> const float* {
        if (LAYER == 1) {
            const int ic  = kt >> 5;
            const int k0r = (kt & 31) * KT;
            return Aq[ic] + (size_t)(m0 + lr + rowoff) * DCOMP + k0r + lc;
        } else {
            return A0 + (size_t)(m0 + lr + rowoff) * KTOT + (size_t)kt * KT + lc;
        }
    };
    auto srcB = [&](int kt, int rowoff) -> const float* {
        const int ic  = kt >> 5;
        const int k0r = (kt & 31) * KT;
        return Wv[c_midx[ic][oc]] + (size_t)(n0r + lr + rowoff) * DCOMP + k0r + lc;
    };

#if HAVE_ASYNC
    auto issueTile = [&](int kt, int buf) {
        async_g2l_16B(srcA(kt, 0),  &As[buf][lr * LDA + lc]);
        async_g2l_16B(srcA(kt, 64), &As[buf][(lr + 64) * LDA + lc]);
        async_g2l_16B(srcB(kt, 0),  &Bs[buf][lr * LDA + lc]);
        async_g2l_16B(srcB(kt, 64), &Bs[buf][(lr + 64) * LDA + lc]);
    };
#else
    v4f ra[2], rb[2];
    auto loadTile = [&](int kt) {
        ra[0] = gload4(srcA(kt, 0));
        ra[1] = gload4(srcA(kt, 64));
        rb[0] = gload4(srcB(kt, 0));
        rb[1] = gload4(srcB(kt, 64));
    };
    auto storeTile = [&](int buf) {
        *(v4f*)&As[buf][lr * LDA + lc]        = ra[0];
        *(v4f*)&As[buf][(lr + 64) * LDA + lc] = ra[1];
        *(v4f*)&Bs[buf][lr * LDA + lc]        = rb[0];
        *(v4f*)&Bs[buf][(lr + 64) * LDA + lc] = rb[1];
    };
#endif

    // WMMA fragment addressing (ISA 32-bit A 16x4 / C-D layouts):
    // lane L holds rows/cols (L&15) and K pair kk = 2*(L>>4).
    const int kk   = (lane >> 4) << 1;
    const int mrow = lane & 15;
    const int am0  = wm * 32;
    const int bn0  = wn * 64;

    // sign folded into A fragments: s*(A)*B == s*(A*B), s uniform per K-chunk
    auto compute = [&](int buf, float s) {
        const float* Ap = As[buf];
        const float* Bp = Bs[buf];
#pragma unroll
        for (int ks = 0; ks < KT / 4; ++ks) {
            const int ko = ks * 4 + kk;
            v2f a0 = *(const v2f*)&Ap[(am0 + mrow) * LDA + ko];
            v2f a1 = *(const v2f*)&Ap[(am0 + 16 + mrow) * LDA + ko];
            a0 *= s;
            a1 *= s;
            v2f b0 = *(const v2f*)&Bp[(bn0 + 0  + mrow) * LDA + ko];
            v2f b1 = *(const v2f*)&Bp[(bn0 + 16 + mrow) * LDA + ko];
            v2f b2 = *(const v2f*)&Bp[(bn0 + 32 + mrow) * LDA + ko];
            v2f b3 = *(const v2f*)&Bp[(bn0 + 48 + mrow) * LDA + ko];
            acc[0][0] = wmma4(a0, b0, acc[0][0]);
            acc[0][1] = wmma4(a0, b1, acc[0][1]);
            acc[0][2] = wmma4(a0, b2, acc[0][2]);
            acc[0][3] = wmma4(a0, b3, acc[0][3]);
            acc[1][0] = wmma4(a1, b0, acc[1][0]);
            acc[1][1] = wmma4(a1, b1, acc[1][1]);
            acc[1][2] = wmma4(a1, b2, acc[1][2]);
            acc[1][3] = wmma4(a1, b3, acc[1][3]);
        }
    };

    // ---- double-buffered main loop (one barrier per K-chunk) ----
#if HAVE_ASYNC
    issueTile(0, 0);
    __builtin_amdgcn_s_wait_asynccnt(0);
#else
    loadTile(0);
    storeTile(0);
#endif
    __syncthreads();
    int cur = 0;
    for (int kt = 0; kt < NKT; ++kt) {
        const int nxt = cur ^ 1;
        if (kt + 2 < NKT) {  // prefetch two chunks ahead (global_prefetch_b8)
            __builtin_prefetch(srcA(kt + 2, 0), 0, 1);
            __builtin_prefetch(srcB(kt + 2, 0), 0, 1);
        }
#if HAVE_ASYNC
        if (kt + 1 < NKT) issueTile(kt + 1, nxt);  // async fill other buffer
        compute(cur, c_sgn[kt >> 5][oc]);
        if (kt + 1 < NKT) __builtin_amdgcn_s_wait_asynccnt(0);
#else
        if (kt + 1 < NKT) loadTile(kt + 1);        // global -> regs
        compute(cur, c_sgn[kt >> 5][oc]);
        if (kt + 1 < NKT) storeTile(nxt);          // regs -> other buffer
#endif
        __syncthreads();
        cur = nxt;
    }

    // ---- epilogue: bias (+ReLU) and store ----
    const int mloc = (lane >> 4) << 3;  // 0 or 8
    const int ncol = lane & 15;
    const float* bias = Bv[oc];
#pragma unroll
    for (int mi = 0; mi < 2; ++mi) {
#pragma unroll
        for (int nj = 0; nj < 4; ++nj) {
            const int   nloc = bn0 + nj * 16 + ncol;  // 0..127 in block
            const float bv   = gloadf(bias + n0r + nloc);
#pragma unroll
            for (int r = 0; r < 8; ++r) {
                const int m = m0 + am0 + mi * 16 + mloc + r;
                float v = acc[mi][nj][r] + bv;
                if (LAYER == 1) {
                    v = fmaxf(v, 0.f);
                    gstoref(Out + (size_t)m * KTOT + (size_t)(oc * DCOMP + n0r + nloc), v);
                } else {
                    gstoref(Out + (size_t)oc * ((size_t)NROWS * DCOMP) +
                                  (size_t)m * DCOMP + (size_t)(n0r + nloc), v);
                }
            }
        }
    }
}

extern "C" void kernel_launch(void* const* d_in, const int* in_sizes, int n_in,
                              void* d_out, int out_size, void* d_ws, size_t ws_size,
                              hipStream_t stream) {
    const float* q[4];
    const float* w1[4];
    const float* w2[4];
    const float* b1[4];
    const float* b2[4];
    for (int c = 0; c < 4; ++c) {
        q[c]  = (const float*)d_in[0 + c];
        w1[c] = (const float*)d_in[4 + c];
        w2[c] = (const float*)d_in[8 + c];
        b1[c] = (const float*)d_in[12 + c];
        b2[c] = (const float*)d_in[16 + c];
    }
    float* h   = (float*)d_ws;   // [NROWS, 2048] fp32 = 256 MiB scratch
    float* out = (float*)d_out;  // 4 components concatenated, each [NROWS, 512]

    dim3 grid(KTOT / N_BLK, NROWS / M_BLK);  // (16, 256)
    qmlp_gemm<1><<<grid, 256, 0, stream>>>(
        q[0], q[1], q[2], q[3],
        w1[0], w1[1], w1[2], w1[3],
        b1[0], b1[1], b1[2], b1[3], h);
    qmlp_gemm<2><<<grid, 256, 0, stream>>>(
        h, h, h, h,
        w2[0], w2[1], w2[2], w2[3],
        b2[0], b2[1], b2[2], b2[3], out);
}